// SelfRouting2d_35828617183258
// MI455X (gfx1250) — compile-verified
//
#include <hip/hip_runtime.h>

// Problem constants (from the reference)
#define BATCH     64
#define IN_UNITS  64      // I
#define N_CAPS    2048    // N
#define NUM_UNITS 32      // K
#define UNIT_SIZE 16      // U
#define JCOLS     (NUM_UNITS * UNIT_SIZE)   // 512 flattened (k,u)
#define NTILE     1024    // n-columns per block (4 per thread, b128 loads)
#define TPAD      (NTILE + 1)  // 1025; 1025 % 64 == 1 -> conflict-free columns

typedef __attribute__((ext_vector_type(2))) float v2f;
typedef __attribute__((ext_vector_type(4))) float v4f;
typedef __attribute__((ext_vector_type(8))) float v8f;

// ---------------------------------------------------------------------------
// Kernel 0: zero the workspace accumulators (y[64*64] and S[64]).
// ---------------------------------------------------------------------------
__global__ void zero_ws_kernel(float* __restrict__ ws, int n) {
  int i = blockIdx.x * blockDim.x + threadIdx.x;
  if (i < n) ws[i] = 0.0f;
}

// ---------------------------------------------------------------------------
// Kernel 1 (memory-bound, single pass over x = 32 MB, HBM floor ~1.4 us):
//   a[b,n]  = sqrt(sum_i x[b,i,n]^2)
//   S[b]   += sum_n a[b,n]
//   y[b,i] += sum_n a[b,n] * x[b,i,n]
// Block = (b, 1024-column chunk). Each thread owns 4 columns and loads them
// as one float4 per i -> global_load_b128, 512B contiguous per wave per i.
// Tile staged in LDS (64 x 1025 floats ~= 256 KB of the 320 KB/WGP) so HBM
// sees each element exactly once; stride 1025 => bank = (i + n) % 64,
// conflict-free for the column-major phase-2 reads.
// ---------------------------------------------------------------------------
__global__ __launch_bounds__(256) void selfroute_reduce_kernel(
    const float* __restrict__ x, float* __restrict__ y, float* __restrict__ S) {
  __shared__ float tile[IN_UNITS * TPAD];
  __shared__ float a_sh[NTILE];
  __shared__ float s_red;

  const int tid = threadIdx.x;
  const int b   = blockIdx.x / (N_CAPS / NTILE);
  const int n0  = (blockIdx.x % (N_CAPS / NTILE)) * NTILE;

  if (tid == 0) s_red = 0.0f;
  __syncthreads();

  // Phase 1: thread owns columns 4*tid .. 4*tid+3 for every i.
  const float* xb = x + (size_t)b * IN_UNITS * N_CAPS + n0 + 4 * tid;
  v4f sq = {};
#pragma unroll
  for (int i = 0; i < IN_UNITS; ++i) {
    v4f v = *reinterpret_cast<const v4f*>(xb + (size_t)i * N_CAPS);
    *reinterpret_cast<v4f*>(&tile[i * TPAD + 4 * tid]) = v;   // ds_store_b128
    sq += v * v;
  }
  v4f a;
  a.x = sqrtf(sq.x); a.y = sqrtf(sq.y); a.z = sqrtf(sq.z); a.w = sqrtf(sq.w);
  *reinterpret_cast<v4f*>(&a_sh[4 * tid]) = a;                // ds_store_b128
  atomicAdd(&s_red, a.x + a.y + a.z + a.w);                   // ds_add_f32
  __syncthreads();
  if (tid == 0) atomicAdd(&S[b], s_red);

  // Phase 2: thread (i = tid&63, quarter q = tid>>6) reduces 256 columns.
  // a_sh[n] is a same-address broadcast across the wave; tile reads are
  // conflict-free thanks to the 1025 stride.
  const int i = tid & (IN_UNITS - 1);
  const int q = tid >> 6;
  const float* trow = &tile[i * TPAD + q * (NTILE / 4)];
  const float* arow = &a_sh[q * (NTILE / 4)];
  float part = 0.0f;
#pragma unroll 8
  for (int t = 0; t < NTILE / 4; ++t) {
    part = fmaf(arow[t], trow[t], part);
  }
  atomicAdd(&y[b * IN_UNITS + i], part);
}

// ---------------------------------------------------------------------------
// Kernel 2: pose[b, j] = (1/S[b]) * sum_i y[b,i] * W1[j,i]   (j = k*16+u)
// C(64x512) = Y(64x64) * W1^T(64x512), fp32 via V_WMMA_F32_16X16X4_F32.
// One wave per 16x16 output tile; 16 chained K=4 WMMAs over I=64.
// A(16x4) layout: lane m = lane&15, half h = lane>>4 holds K = {2h, 2h+1}.
// B(4x16)/D(16x16): rows striped across lanes within a VGPR.
// Epilogue scales with v_rcp_f32 (no IEEE divide sequences needed).
// ---------------------------------------------------------------------------
__global__ __launch_bounds__(256) void selfroute_gemm_wmma_kernel(
    const float* __restrict__ y, const float* __restrict__ W1,
    const float* __restrict__ S, float* __restrict__ out) {
  const int lane = threadIdx.x & 31;
  const int wid  = (blockIdx.x * blockDim.x + threadIdx.x) >> 5;  // 0..127
  const int mt   = wid & 3;        // BATCH/16  = 4 tiles
  const int nt   = wid >> 2;       // JCOLS/16  = 32 tiles
  const int m0   = mt * 16;
  const int n0   = nt * 16;
  const int l15  = lane & 15;
  const int h    = lane >> 4;

  const float* yrow = y  + (size_t)(m0 + l15) * IN_UNITS;
  const float* wrow = W1 + (size_t)(n0 + l15) * IN_UNITS;

  v8f c = {};
#pragma unroll
  for (int kk = 0; kk < IN_UNITS / 4; ++kk) {
    const int k0 = 4 * kk + 2 * h;
    v2f av = *reinterpret_cast<const v2f*>(yrow + k0);   // A: Y[m, k0..k0+1]
    v2f bv = *reinterpret_cast<const v2f*>(wrow + k0);   // B: W1[j, k0..k0+1]
    // 8 args: (neg_a, A, neg_b, B, c_mod, C, reuse_a, reuse_b)
    c = __builtin_amdgcn_wmma_f32_16x16x4_f32(
        false, av, false, bv, (short)0, c, false, false);
  }

  // Epilogue: D VGPR v holds row m0 + 8h + v, col n0 + l15. Scale by 1/S_b
  // using the fast reciprocal (v_rcp_f32) instead of a full IEEE divide.
#pragma unroll
  for (int v = 0; v < 8; ++v) {
    const int row  = m0 + 8 * h + v;
    const float is = __builtin_amdgcn_rcpf(S[row]);
    out[(size_t)row * JCOLS + n0 + l15] = c[v] * is;
  }
}

// ---------------------------------------------------------------------------
// Launcher. Inputs: x, W1, W2, b2. W2/b2 provably cancel: logit is constant
// along the softmax axis (K), so cij == 1/K uniformly, which then cancels in
// the coeff normalization over n.
// ---------------------------------------------------------------------------
extern "C" void kernel_launch(void* const* d_in, const int* in_sizes, int n_in,
                              void* d_out, int out_size, void* d_ws, size_t ws_size,
                              hipStream_t stream) {
  const float* x  = (const float*)d_in[0];
  const float* W1 = (const float*)d_in[1];
  // d_in[2] (W2) and d_in[3] (b2) are mathematically unused — see analysis.
  float* out = (float*)d_out;

  float* y = (float*)d_ws;                 // BATCH * IN_UNITS floats
  float* S = y + BATCH * IN_UNITS;         // BATCH floats

  const int nzero = BATCH * IN_UNITS + BATCH;
  zero_ws_kernel<<<(nzero + 255) / 256, 256, 0, stream>>>((float*)d_ws, nzero);

  selfroute_reduce_kernel<<<BATCH * (N_CAPS / NTILE), 256, 0, stream>>>(x, y, S);

  // 128 waves total (4 x 32 tiles), 8 waves per 256-thread block -> 16 blocks.
  selfroute_gemm_wmma_kernel<<<16, 256, 0, stream>>>(y, W1, S, out);
}